// Fusion_62878321213589
// MI455X (gfx1250) — compile-verified
//
#include <hip/hip_runtime.h>
#include <math.h>

// ---------------- problem constants ----------------
#define BB 32
#define CC 32
#define LL 512
#define INTRA 1024
#define INTER 256
#define DCP 64
#define DDE 512
// 1/sqrt(512) * SCALE(0.05)
#define NEG_SCALE (0.044194173824159216f * 0.05f)

typedef __attribute__((ext_vector_type(2))) float v2f;
typedef __attribute__((ext_vector_type(8))) float v8f;

static __device__ __forceinline__ v8f wmma_f32(v2f a, v2f b, v8f c) {
  // V_WMMA_F32_16X16X4_F32 : D = A(16x4) * B(4x16) + C(16x16)
  return __builtin_amdgcn_wmma_f32_16x16x4_f32(
      /*neg_a=*/false, a, /*neg_b=*/false, b,
      /*c_mod=*/(short)0, c, /*reuse_a=*/false, /*reuse_b=*/false);
}

static __device__ __forceinline__ float sigmoidf_(float x) {
  return 1.0f / (1.0f + __expf(-x));
}

// =====================================================================
// Kernel 1: fused compress
//   hidden = relu(map_raw @ cW1^T + cb1)        (C*INTRA, DCP) K=INTER
//   out    = hidden @ cW2^T + cb2               (C*INTRA, DCP) K=DCP
//   stored TRANSPOSED: map_tmp[c][dcp][intra]
// grid: 256 blocks x 256 thr (8 waves, each wave one 16-row strip x 64 cols)
// =====================================================================
__global__ __launch_bounds__(256) void compress_kernel(
    const float* __restrict__ map_raw, const float* __restrict__ cW1,
    const float* __restrict__ cb1, const float* __restrict__ cW2,
    const float* __restrict__ cb2, float* __restrict__ map_tmp) {
  __shared__ float s_hid[8][16 * 66];  // per-wave 16x64 tile, stride 66
  const int t = threadIdx.x;
  const int w = t >> 5;
  const int lane = t & 31;
  const int ln = lane & 15;
  const int half = lane >> 4;
  const int m0 = (blockIdx.x * 8 + w) * 16;  // row in (C*INTRA)
  const int c = m0 >> 10;                    // /INTRA

  // ---- GEMM1: K = INTER = 256 ----
  const float* Arow = map_raw + (size_t)(m0 + ln) * INTER;
  v8f acc[4] = {};
#pragma unroll 4
  for (int k = 0; k < INTER; k += 4) {
    if ((k & 63) == 0) __builtin_prefetch(Arow + k + 64, 0, 1);
    const int kk = k + 2 * half;
    v2f a;
    a.x = Arow[kk];
    a.y = Arow[kk + 1];
#pragma unroll
    for (int nt = 0; nt < 4; ++nt) {
      const int n = nt * 16 + ln;
      v2f b;
      b.x = cW1[n * INTER + kk];
      b.y = cW1[n * INTER + kk + 1];
      acc[nt] = wmma_f32(a, b, acc[nt]);
    }
  }
  // relu + bias, park 16x64 hidden tile in LDS (D-layout -> A-layout shuffle)
#pragma unroll
  for (int nt = 0; nt < 4; ++nt) {
    const int n = nt * 16 + ln;
    const float bias = cb1[n];
#pragma unroll
    for (int r = 0; r < 8; ++r) {
      float v = acc[nt][r] + bias;
      s_hid[w][(r + 8 * half) * 66 + n] = v > 0.0f ? v : 0.0f;
    }
  }
  __syncthreads();

  // ---- GEMM2: K = DCP = 64 ----
  v8f acc2[4] = {};
#pragma unroll
  for (int k = 0; k < DCP; k += 4) {
    const int kk = k + 2 * half;
    v2f a;
    a.x = s_hid[w][ln * 66 + kk];
    a.y = s_hid[w][ln * 66 + kk + 1];
#pragma unroll
    for (int nt = 0; nt < 4; ++nt) {
      const int n = nt * 16 + ln;
      v2f b;
      b.x = cW2[n * DCP + kk];
      b.y = cW2[n * DCP + kk + 1];
      acc2[nt] = wmma_f32(a, b, acc2[nt]);
    }
  }
  // bias + transposed store: map_tmp[c][dcp][intra]
#pragma unroll
  for (int nt = 0; nt < 4; ++nt) {
    const int dcp = nt * 16 + ln;
    const float bias = cb2[dcp];
    float* dst = map_tmp + ((size_t)c * DCP + dcp) * INTRA;
#pragma unroll
    for (int r = 0; r < 8; ++r) {
      const int m = m0 + r + 8 * half;
      dst[m & (INTRA - 1)] = acc2[nt][r] + bias;
    }
  }
}

// =====================================================================
// Kernel 2: denoise linear 1
//   H = relu(map_tmp(2048x1024) @ dW1^T(1024x512) + db1)  -> (2048x512)
// grid (16, 8) x 256 thr; B tile 64x64 double-buffered in LDS
// =====================================================================
__global__ __launch_bounds__(256) void denoise1_kernel(
    const float* __restrict__ A, const float* __restrict__ dW1,
    const float* __restrict__ db1, float* __restrict__ Hout) {
  __shared__ float sB[2][64 * 68];  // stride 68: conflict-free, f4-aligned
  const int t = threadIdx.x;
  const int w = t >> 5;
  const int lane = t & 31;
  const int ln = lane & 15;
  const int half = lane >> 4;
  const int m0 = (blockIdx.x * 8 + w) * 16;
  const int n0 = blockIdx.y * 64;
  const float* Arow = A + (size_t)(m0 + ln) * INTRA;

  // cooperative stage indices: 64 rows x 16 float4 / 256 thr = 4 each
  const int sn = t >> 4;        // row 0..15 base (x4 below)
  const int skq = t & 15;       // float4 index within row
  v8f acc[4] = {};

  // prologue: stage k0 = 0 into buffer 0
#pragma unroll
  for (int i = 0; i < 4; ++i) {
    const int n = sn + i * 16;
    *(float4*)&sB[0][n * 68 + skq * 4] =
        *(const float4*)(dW1 + (size_t)(n0 + n) * INTRA + skq * 4);
  }
  __syncthreads();

  int buf = 0;
  for (int k0 = 0; k0 < INTRA; k0 += 64) {
    if (k0 + 64 < INTRA) {  // stream next B tile into the other buffer
#pragma unroll
      for (int i = 0; i < 4; ++i) {
        const int n = sn + i * 16;
        *(float4*)&sB[buf ^ 1][n * 68 + skq * 4] =
            *(const float4*)(dW1 + (size_t)(n0 + n) * INTRA + (k0 + 64) +
                             skq * 4);
      }
    }
    __builtin_prefetch(Arow + k0 + 64, 0, 1);  // global_prefetch for A stream
#pragma unroll 4
    for (int k = 0; k < 64; k += 4) {
      const int kk = k + 2 * half;
      v2f a;
      a.x = Arow[k0 + kk];
      a.y = Arow[k0 + kk + 1];
#pragma unroll
      for (int nt = 0; nt < 4; ++nt) {
        const int n = nt * 16 + ln;
        v2f b;
        b.x = sB[buf][n * 68 + kk];
        b.y = sB[buf][n * 68 + kk + 1];
        acc[nt] = wmma_f32(a, b, acc[nt]);
      }
    }
    __syncthreads();
    buf ^= 1;
  }
#pragma unroll
  for (int nt = 0; nt < 4; ++nt) {
    const int n = n0 + nt * 16 + ln;
    const float bias = db1[n];
#pragma unroll
    for (int r = 0; r < 8; ++r) {
      const int m = m0 + r + 8 * half;
      float v = acc[nt][r] + bias;
      Hout[(size_t)m * DDE + n] = v > 0.0f ? v : 0.0f;
    }
  }
}

// =====================================================================
// Kernel 3: denoise linear 2
//   map_st = H(2048x512) @ dW2^T(512x1024) + db2
//   stored TRANSPOSED: map_stT[c][intra][dcp]  (gathered column = 256B)
// grid (16, 16) x 256 thr; B tile 64x64 double-buffered in LDS
// =====================================================================
__global__ __launch_bounds__(256) void denoise2_kernel(
    const float* __restrict__ H, const float* __restrict__ dW2,
    const float* __restrict__ db2, float* __restrict__ map_stT) {
  __shared__ float sB[2][64 * 68];
  const int t = threadIdx.x;
  const int w = t >> 5;
  const int lane = t & 31;
  const int ln = lane & 15;
  const int half = lane >> 4;
  const int m0 = (blockIdx.x * 8 + w) * 16;
  const int n0 = blockIdx.y * 64;
  const float* Arow = H + (size_t)(m0 + ln) * DDE;

  const int sn = t >> 4;
  const int skq = t & 15;
  v8f acc[4] = {};

#pragma unroll
  for (int i = 0; i < 4; ++i) {
    const int n = sn + i * 16;
    *(float4*)&sB[0][n * 68 + skq * 4] =
        *(const float4*)(dW2 + (size_t)(n0 + n) * DDE + skq * 4);
  }
  __syncthreads();

  int buf = 0;
  for (int k0 = 0; k0 < DDE; k0 += 64) {
    if (k0 + 64 < DDE) {
#pragma unroll
      for (int i = 0; i < 4; ++i) {
        const int n = sn + i * 16;
        *(float4*)&sB[buf ^ 1][n * 68 + skq * 4] =
            *(const float4*)(dW2 + (size_t)(n0 + n) * DDE + (k0 + 64) +
                             skq * 4);
      }
    }
    __builtin_prefetch(Arow + k0 + 64, 0, 1);
#pragma unroll 4
    for (int k = 0; k < 64; k += 4) {
      const int kk = k + 2 * half;
      v2f a;
      a.x = Arow[k0 + kk];
      a.y = Arow[k0 + kk + 1];
#pragma unroll
      for (int nt = 0; nt < 4; ++nt) {
        const int n = nt * 16 + ln;
        v2f b;
        b.x = sB[buf][n * 68 + kk];
        b.y = sB[buf][n * 68 + kk + 1];
        acc[nt] = wmma_f32(a, b, acc[nt]);
      }
    }
    __syncthreads();
    buf ^= 1;
  }
#pragma unroll
  for (int nt = 0; nt < 4; ++nt) {
    const int intra = n0 + nt * 16 + ln;
    const float bias = db2[intra];
#pragma unroll
    for (int r = 0; r < 8; ++r) {
      const int m = m0 + r + 8 * half;  // (c, dcp)
      const int ch = m >> 6;
      const int dcp = m & 63;
      map_stT[((size_t)ch * INTRA + intra) * DCP + dcp] = acc[nt][r] + bias;
    }
  }
}

// =====================================================================
// Kernel 4: fused gather + L1 score + softmax + recombine + lamda blend
// one block per (b,c); map_stT bank (8 MB) is L2-resident on MI455X
// =====================================================================
__global__ __launch_bounds__(256) void score_kernel(
    const float* __restrict__ x_loc, const int* __restrict__ indices,
    const float* __restrict__ map_stT, const float* __restrict__ lamda1,
    const float* __restrict__ lamda2, float* __restrict__ out) {
  __shared__ int s_idx[LL];
  __shared__ float s_x[LL];
  __shared__ float s_part[256];
  __shared__ float s_logit[DCP];
  __shared__ float s_w[DCP];

  const int t = threadIdx.x;
  const int b = blockIdx.x >> 5;  // /C
  const int c = blockIdx.x & 31;
  const float* xrow = x_loc + ((size_t)b * CC + c) * LL;
  const int* irow = indices + (size_t)b * LL;
  for (int l = t; l < LL; l += 256) {
    s_idx[l] = irow[l];
    s_x[l] = xrow[l];
  }
  __syncthreads();

  // L1 distances: group g (of 4) covers 128 l's, thread within group owns d
  const int g = t >> 6;
  const int d = t & 63;
  const float* base_c = map_stT + (size_t)c * INTRA * DCP;
  float acc = 0.0f;
  for (int j = 0; j < 128; ++j) {
    const int l = g * 128 + j;
    const float v = base_c[(size_t)s_idx[l] * DCP + d];
    acc += fabsf(s_x[l] - v);
  }
  s_part[t] = acc;
  __syncthreads();

  if (t < DCP) {
    const float s =
        s_part[t] + s_part[64 + t] + s_part[128 + t] + s_part[192 + t];
    s_logit[t] = -s * NEG_SCALE;  // (-sum/sqrt(L)) * SCALE
  }
  __syncthreads();
  if (t < DCP) {
    float mx = -1e30f;
    for (int i = 0; i < DCP; ++i) mx = fmaxf(mx, s_logit[i]);
    float sum = 0.0f;
    for (int i = 0; i < DCP; ++i) sum += __expf(s_logit[i] - mx);
    s_w[t] = __expf(s_logit[t] - mx) / sum;
  }
  __syncthreads();

  const float lam_c = sigmoidf_(lamda1[c]);
  float* orow = out + ((size_t)b * CC + c) * LL;
  for (int l = t; l < LL; l += 256) {
    const float* col = base_c + (size_t)s_idx[l] * DCP;
    float dot = 0.0f;
#pragma unroll
    for (int i = 0; i < DCP / 4; ++i) {
      const float4 wv = *(const float4*)(s_w + i * 4);
      const float4 cv = *(const float4*)(col + i * 4);
      dot += wv.x * cv.x + wv.y * cv.y + wv.z * cv.z + wv.w * cv.w;
    }
    const float lam = lam_c * sigmoidf_(lamda2[l]);
    orow[l] = dot * lam + s_x[l] * (1.0f - lam);
  }
}

// =====================================================================
extern "C" void kernel_launch(void* const* d_in, const int* in_sizes, int n_in,
                              void* d_out, int out_size, void* d_ws,
                              size_t ws_size, hipStream_t stream) {
  (void)in_sizes;
  (void)n_in;
  (void)out_size;
  (void)ws_size;
  const float* x_loc = (const float*)d_in[0];
  const int* indices = (const int*)d_in[1];
  const float* map_raw = (const float*)d_in[2];
  const float* cW1 = (const float*)d_in[3];
  const float* cb1 = (const float*)d_in[4];
  const float* cW2 = (const float*)d_in[5];
  const float* cb2 = (const float*)d_in[6];
  const float* dW1 = (const float*)d_in[7];
  const float* db1 = (const float*)d_in[8];
  const float* dW2 = (const float*)d_in[9];
  const float* db2 = (const float*)d_in[10];
  const float* lamda1 = (const float*)d_in[11];
  const float* lamda2 = (const float*)d_in[12];
  float* out = (float*)d_out;

  // workspace layout (floats): map_tmp 2M | hidden2 1M | map_stT 2M = 20 MB
  float* map_tmp = (float*)d_ws;
  float* hidden2 = map_tmp + (size_t)CC * DCP * INTRA;
  float* map_stT = hidden2 + (size_t)(CC * DCP) * DDE;

  compress_kernel<<<256, 256, 0, stream>>>(map_raw, cW1, cb1, cW2, cb2,
                                           map_tmp);
  denoise1_kernel<<<dim3(16, 8), 256, 0, stream>>>(map_tmp, dW1, db1, hidden2);
  denoise2_kernel<<<dim3(16, 16), 256, 0, stream>>>(hidden2, dW2, db2,
                                                    map_stT);
  score_kernel<<<BB * CC, 256, 0, stream>>>(x_loc, indices, map_stT, lamda1,
                                            lamda2, out);
}